// RealSymplecticReactionDiffusion2D_84971632984247
// MI455X (gfx1250) — compile-verified
//
#include <hip/hip_runtime.h>
#include <math.h>

// ---------------------------------------------------------------------------
// Problem constants (from reference setup_inputs)
// ---------------------------------------------------------------------------
#define BATCH   4
#define NTOK    4096       // N  (= 64*64)
#define DMODEL  1024       // d
#define HW      64         // plane side
#define PLANE   (HW*HW)    // 4096 cells per (b,c) plane
#define KDIM    (2*DMODEL) // 2048 contraction dim of output GEMM

typedef float v2f __attribute__((ext_vector_type(2)));
typedef float v8f __attribute__((ext_vector_type(8)));
typedef float v4f __attribute__((ext_vector_type(4)));

__device__ __forceinline__ float softplusf(float x) {
    return (x > 20.0f) ? x : log1pf(expf(x));
}

// ---------------------------------------------------------------------------
// 1) pooled[b,c] = sum_n x[b,n,c]   (mean applied later)
// ---------------------------------------------------------------------------
__global__ __launch_bounds__(256)
void pool_sum_kernel(const float* __restrict__ x, float* __restrict__ pooled) {
    const int c     = blockIdx.x * 256 + threadIdx.x;
    const int b     = blockIdx.y;
    const int chunk = blockIdx.z;                  // 16 chunks of 256 rows
    const float* xp = x + ((size_t)b * NTOK + (size_t)chunk * 256) * DMODEL + c;
    float s = 0.0f;
    #pragma unroll 8
    for (int n = 0; n < 256; ++n) s += xp[(size_t)n * DMODEL];
    atomicAdd(&pooled[b * DMODEL + c], s);
}

// ---------------------------------------------------------------------------
// 2) transpose x (B,N,d) -> xt (B,d,N) so PDE planes are contiguous.
// ---------------------------------------------------------------------------
__global__ __launch_bounds__(256)
void transpose_kernel(const float* __restrict__ x, float* __restrict__ xt) {
    __shared__ float tile[32][33];
    const int b  = blockIdx.z;
    const int d0 = blockIdx.x * 32;
    const int n0 = blockIdx.y * 32;
    const int col = threadIdx.x;                   // 0..31
    for (int r = threadIdx.y; r < 32; r += 8)
        tile[r][col] = x[((size_t)b * NTOK + n0 + r) * DMODEL + d0 + col];
    __syncthreads();
    for (int r = threadIdx.y; r < 32; r += 8)
        xt[((size_t)b * DMODEL + d0 + r) * NTOK + n0 + col] = tile[col][r];
}

// ---------------------------------------------------------------------------
// 3a) per-channel reaction params
// ---------------------------------------------------------------------------
__global__ __launch_bounds__(256)
void chan_param_kernel(const float* __restrict__ alpha_raw,
                       const float* __restrict__ beta_raw,
                       float* __restrict__ chan) {
    const int c = blockIdx.x * 256 + threadIdx.x;
    if (c < DMODEL) {
        chan[c]          = 0.2f * tanhf(alpha_raw[c]);
        chan[DMODEL + c] = softplusf(beta_raw[c]) + 1e-4f;
    }
}

// ---------------------------------------------------------------------------
// 3b) gamma[b,j] = min(softplus(dot(pooled[b]/N, Wg[j]) + bg[j] + ldb[j]), 0.25)
// ---------------------------------------------------------------------------
__global__ __launch_bounds__(256)
void gamma_kernel(const float* __restrict__ pooled, const float* __restrict__ Wg,
                  const float* __restrict__ bg, const float* __restrict__ ldb,
                  float* __restrict__ gamma) {
    const int id = blockIdx.x * 256 + threadIdx.x;   // 0 .. B*d-1
    const int b  = id >> 10;
    const int j  = id & (DMODEL - 1);
    const float* p = pooled + b * DMODEL;
    const v4f* w   = (const v4f*)(Wg + (size_t)j * DMODEL);
    float acc = 0.0f;
    #pragma unroll 4
    for (int k = 0; k < DMODEL / 4; ++k) {
        v4f wv = w[k];
        acc += wv.x * p[4*k+0] + wv.y * p[4*k+1] + wv.z * p[4*k+2] + wv.w * p[4*k+3];
    }
    acc *= (1.0f / (float)NTOK);                     // pooled was a sum -> mean
    float g = softplusf(acc + bg[j] + ldb[j]);
    gamma[id] = fminf(g, 0.25f);
}

// ---------------------------------------------------------------------------
// 4) PDE kernel: one workgroup per (b,c) plane, u/v resident in LDS (32 KB).
// ---------------------------------------------------------------------------
__device__ __forceinline__ void lap_phase(const float* __restrict__ S, int tid,
                                          float lp[16]) {
    #pragma unroll
    for (int ii = 0; ii < 16; ++ii) {
        const int cell = tid + 256 * ii;
        const int y = cell >> 6, xx = cell & 63;
        lp[ii] = S[(((y + 1) & 63) << 6) + xx]
               + S[(((y + 63) & 63) << 6) + xx]
               + S[(y << 6) + ((xx + 1) & 63)]
               + S[(y << 6) + ((xx + 63) & 63)]
               - 4.0f * S[cell];
    }
}

__device__ __forceinline__ void react_phase(float* __restrict__ U, float* __restrict__ V,
                                            int tid, float a, float bb, float E,
                                            float hdt, bool smallA) {
    #pragma unroll
    for (int ii = 0; ii < 16; ++ii) {
        const int cell = tid + 256 * ii;
        float u = U[cell], v = V[cell];
        float q = u * u + v * v;
        float denom = bb * q + (a - bb * q) * E;
        float qg = a * q / fmaxf(denom, 1e-8f);
        float qz = q / (1.0f + bb * q * hdt);
        float qn = fmaxf(smallA ? qz : qg, 0.0f);
        float sc = sqrtf(qn / fmaxf(q, 1e-8f));
        U[cell] = u * sc;
        V[cell] = v * sc;
    }
}

__global__ __launch_bounds__(256)
void pde_kernel(const float* __restrict__ xt, const float* __restrict__ gamma,
                const float* __restrict__ chan, const int* __restrict__ kptr,
                float* __restrict__ ut, float* __restrict__ vt) {
    __shared__ float U[PLANE];
    __shared__ float V[PLANE];
    const int bc  = blockIdx.x;            // b*1024 + c
    const int c   = bc & (DMODEL - 1);
    const int tid = threadIdx.x;

    const int ks  = max(kptr[0], 1);
    const float dt  = 1.0f / (float)ks;
    const float hdt = 0.5f * dt;
    const float alpha = chan[c];
    const float beta  = chan[DMODEL + c];
    const float g   = gamma[bc];
    const float a   = 2.0f * alpha;
    const float bb  = 2.0f * beta;
    const float E   = expf(-a * hdt);
    const bool smallA = fabsf(a) < 1e-6f;
    const float c1 = 0.5f * dt * g;
    const float c2 = dt * g;

    const float* src = xt + (size_t)bc * PLANE;
    for (int i = tid; i < PLANE; i += 256) { U[i] = src[i]; V[i] = 0.0f; }

    float lp[16];
    for (int s = 0; s < ks; ++s) {
        react_phase(U, V, tid, a, bb, E, hdt, smallA);
        __syncthreads();
        lap_phase(U, tid, lp);                                  // v_half
        #pragma unroll
        for (int ii = 0; ii < 16; ++ii) V[tid + 256 * ii] -= c1 * lp[ii];
        __syncthreads();
        lap_phase(V, tid, lp);                                  // u_new
        #pragma unroll
        for (int ii = 0; ii < 16; ++ii) U[tid + 256 * ii] += c2 * lp[ii];
        __syncthreads();
        lap_phase(U, tid, lp);                                  // v_new
        #pragma unroll
        for (int ii = 0; ii < 16; ++ii) V[tid + 256 * ii] -= c1 * lp[ii];
        __syncthreads();
        react_phase(U, V, tid, a, bb, E, hdt, smallA);
    }

    float* du = ut + (size_t)bc * PLANE;
    float* dv = vt + (size_t)bc * PLANE;
    for (int i = tid; i < PLANE; i += 256) { du[i] = U[i]; dv[i] = V[i]; }
}

// ---------------------------------------------------------------------------
// 5) Output GEMM with WMMA f32 16x16x4.
//    Workgroup tile 128(M) x 128(N), 8 waves as 4(M) x 2(N); each wave owns
//    32x64 = 8 accumulator tiles (2 A frags x 4 B frags per k-step).
//    LDS is pair-interleaved [k/2][row][k&1] so every 16x4 fragment is one
//    aligned ds_load_b64.  Double-buffered, register-staged software pipeline
//    overlaps next-stage HBM loads with the current stage's 32 WMMAs; one
//    barrier per K-stage.
// ---------------------------------------------------------------------------
#define GTM 128
#define GTN 128
#define NSTAGE (KDIM / 16)   // 128 K-stages of 16

__global__ __launch_bounds__(256)
void out_gemm_kernel(const float* __restrict__ ut, const float* __restrict__ vt,
                     const float* __restrict__ Wout, const float* __restrict__ x,
                     const float* __restrict__ Dvec, float* __restrict__ out) {
    // [buf][kpair][row][k&1] -> fragment pair contiguous (8B) per row
    __shared__ float As[2][8][GTM][2];
    __shared__ float Bs[2][8][GTN][2];

    const int tid  = threadIdx.x;
    const int wave = tid >> 5;
    const int lane = tid & 31;
    const int half = lane >> 4;     // 0: lanes 0-15, 1: lanes 16-31
    const int lm   = lane & 15;
    const int wm   = wave & 3;      // wave M index (0..3) -> rows wm*32
    const int wn   = wave >> 2;     // wave N index (0..1) -> cols wn*64

    const int m0 = blockIdx.y * GTM;        // global row in (B*N)
    const int j0 = blockIdx.x * GTN;
    const int b  = m0 >> 12;                // 4096 rows per batch; GTM | 4096
    const int n0 = m0 & (NTOK - 1);

    const float* Ubase = ut + (size_t)b * DMODEL * PLANE;
    const float* Vbase = vt + (size_t)b * DMODEL * PLANE;

    v8f acc[2][4];
    #pragma unroll
    for (int t = 0; t < 2; ++t)
        #pragma unroll
        for (int u = 0; u < 4; ++u)
            #pragma unroll
            for (int r = 0; r < 8; ++r) acc[t][u][r] = 0.0f;

    // staging maps: 2048 elements each for A and B, 8 per thread
    const int am   = tid & 127;             // A: m within tile
    const int akk0 = (tid >> 7) * 8;        // A: kk base (0 or 8)
    const int bj   = tid >> 1;              // B: j within tile (0..127)
    const int bkk0 = (tid & 1) * 8;         // B: kk base (0 or 8)

    float aReg[8], bReg[8];

    #define FETCH_STAGE(S)                                                     \
    do {                                                                       \
        const int k0_ = (S) * 16;                                              \
        const bool isU_ = (k0_ < DMODEL);       /* uniform */                  \
        const float* Ab_ = isU_ ? Ubase : Vbase;                               \
        const int kadj_ = isU_ ? k0_ : (k0_ - DMODEL);                         \
        _Pragma("unroll")                                                      \
        for (int i = 0; i < 8; ++i)                                            \
            aReg[i] = Ab_[(size_t)(kadj_ + akk0 + i) * PLANE + n0 + am];       \
        const v4f* wp_ = (const v4f*)(Wout + (size_t)(j0 + bj) * KDIM + k0_ + bkk0); \
        v4f w0_ = wp_[0], w1_ = wp_[1];                                        \
        bReg[0] = w0_.x; bReg[1] = w0_.y; bReg[2] = w0_.z; bReg[3] = w0_.w;    \
        bReg[4] = w1_.x; bReg[5] = w1_.y; bReg[6] = w1_.z; bReg[7] = w1_.w;    \
    } while (0)

    #define STORE_STAGE(BUF)                                                   \
    do {                                                                       \
        _Pragma("unroll")                                                      \
        for (int i = 0; i < 8; ++i) {                                          \
            const int kkA_ = akk0 + i;                                         \
            As[BUF][kkA_ >> 1][am][kkA_ & 1] = aReg[i];                        \
            const int kkB_ = bkk0 + i;                                         \
            Bs[BUF][kkB_ >> 1][bj][kkB_ & 1] = bReg[i];                        \
        }                                                                      \
    } while (0)

    FETCH_STAGE(0);
    STORE_STAGE(0);

    for (int s = 0; s < NSTAGE; ++s) {
        __syncthreads();                 // buf[s&1] ready; prior compute done
        const int cur = s & 1;
        const bool more = (s + 1 < NSTAGE);
        if (more) FETCH_STAGE(s + 1);    // HBM loads overlap the WMMAs below

        #pragma unroll
        for (int kstep = 0; kstep < 4; ++kstep) {
            const int k2 = kstep * 2 + half;   // pair index for this half-wave
            v2f af[2], bf[4];
            af[0] = *(const v2f*)&As[cur][k2][wm * 32 + lm][0];
            af[1] = *(const v2f*)&As[cur][k2][wm * 32 + 16 + lm][0];
            #pragma unroll
            for (int u = 0; u < 4; ++u)
                bf[u] = *(const v2f*)&Bs[cur][k2][wn * 64 + u * 16 + lm][0];
            #pragma unroll
            for (int t = 0; t < 2; ++t)
                #pragma unroll
                for (int u = 0; u < 4; ++u)
                    acc[t][u] = __builtin_amdgcn_wmma_f32_16x16x4_f32(
                        /*neg_a=*/false, af[t], /*neg_b=*/false, bf[u],
                        /*c_mod=*/(short)0, acc[t][u],
                        /*reuse_a=*/false, /*reuse_b=*/false);
        }

        if (more) STORE_STAGE((s + 1) & 1);  // safe: top barrier proved the
                                             // other buffer's readers finished
    }

    // ---- epilogue: out = acc + x*D (fused residual) ----
    // C/D layout: VGPR r -> M = r (lanes 0-15) / r+8 (lanes 16-31), N = lm
    #pragma unroll
    for (int t = 0; t < 2; ++t) {
        #pragma unroll
        for (int u = 0; u < 4; ++u) {
            const int j = j0 + wn * 64 + u * 16 + lm;
            const float dj = Dvec[j];
            #pragma unroll
            for (int r = 0; r < 8; ++r) {
                const int m = m0 + wm * 32 + t * 16 + r + half * 8;
                const size_t oidx = (size_t)m * DMODEL + j;
                out[oidx] = acc[t][u][r] + x[oidx] * dj;
            }
        }
    }
    #undef FETCH_STAGE
    #undef STORE_STAGE
}

// ---------------------------------------------------------------------------
// launch
// ---------------------------------------------------------------------------
extern "C" void kernel_launch(void* const* d_in, const int* in_sizes, int n_in,
                              void* d_out, int out_size, void* d_ws, size_t ws_size,
                              hipStream_t stream) {
    const float* x         = (const float*)d_in[0];
    const float* Wg        = (const float*)d_in[1];
    const float* bg        = (const float*)d_in[2];
    const float* ldb       = (const float*)d_in[3];
    const float* alpha_raw = (const float*)d_in[4];
    const float* beta_raw  = (const float*)d_in[5];
    const float* Wout      = (const float*)d_in[6];
    const float* Dvec      = (const float*)d_in[7];
    const int*   kptr      = (const int*)d_in[8];
    float* out = (float*)d_out;

    float* ws = (float*)d_ws;
    size_t off = 0;
    float* pooled = ws + off; off += (size_t)BATCH * DMODEL;
    float* gamma  = ws + off; off += (size_t)BATCH * DMODEL;
    float* chan   = ws + off; off += (size_t)2 * DMODEL;
    float* xt     = ws + off; off += (size_t)BATCH * DMODEL * PLANE;
    float* ut     = ws + off; off += (size_t)BATCH * DMODEL * PLANE;
    float* vt     = ws + off; off += (size_t)BATCH * DMODEL * PLANE;

    hipMemsetAsync(pooled, 0, (size_t)BATCH * DMODEL * sizeof(float), stream);

    pool_sum_kernel<<<dim3(DMODEL / 256, BATCH, 16), 256, 0, stream>>>(x, pooled);
    transpose_kernel<<<dim3(DMODEL / 32, NTOK / 32, BATCH), dim3(32, 8), 0, stream>>>(x, xt);
    chan_param_kernel<<<DMODEL / 256, 256, 0, stream>>>(alpha_raw, beta_raw, chan);
    gamma_kernel<<<(BATCH * DMODEL) / 256, 256, 0, stream>>>(pooled, Wg, bg, ldb, gamma);
    pde_kernel<<<BATCH * DMODEL, 256, 0, stream>>>(xt, gamma, chan, kptr, ut, vt);
    out_gemm_kernel<<<dim3(DMODEL / GTN, (BATCH * NTOK) / GTM), 256, 0, stream>>>(
        ut, vt, Wout, x, Dvec, out);
}